// Block_65661460021475
// MI455X (gfx1250) — compile-verified
//
#include <hip/hip_runtime.h>

// ---------------------------------------------------------------------------
// Sizes fixed by the reference.
// ---------------------------------------------------------------------------
#define D_CH   128          // BLOCK (channels)
#define NFRM   65536        // FRAMES
#define NWIN   512          // WINDOW
#define CHUNK  256          // frames per local-IIR chunk
#define NCHUNK (NFRM / CHUNK)   // 256

typedef __attribute__((ext_vector_type(2))) float v2f;
typedef __attribute__((ext_vector_type(4))) float v4f;
typedef __attribute__((ext_vector_type(8))) float v8f;

__device__ __forceinline__ float sigmoidf_(float x) {
    return 1.0f / (1.0f + __expf(-x));
}

// ---------------------------------------------------------------------------
// 0) Per-channel scalars: dec = 0.5 + 0.5*sigmoid(decays) (+1e-12 as in ref),
//    log(dec) for the carry fix-up, gain = 5*sigmoid(gains).
// ---------------------------------------------------------------------------
__global__ void k_prep(const float* __restrict__ decays,
                       const float* __restrict__ gains,
                       float* __restrict__ ws) {
    int c = threadIdx.x;
    if (c < D_CH) {
        float d = 0.5f + sigmoidf_(decays[c]) * 0.5f + 1e-12f;
        ws[c]            = d;            // dec
        ws[D_CH + c]     = __logf(d);    // log(dec)
        ws[2 * D_CH + c] = sigmoidf_(gains[c]) * 5.0f;   // gain
    }
}

// ---------------------------------------------------------------------------
// 1) h = w1 @ relu(cp).  M=128, K=128, N=65536.  f32 WMMA 16x16x4.
//    One wave per 16x16 tile; 8 waves/block cover all M for one N tile
//    so the activation (B) tile is reused out of L0/L2.
// ---------------------------------------------------------------------------
__global__ void k_gemm1(const float* __restrict__ cp,
                        const float* __restrict__ w1,
                        float* __restrict__ h) {
    const int wid  = threadIdx.x >> 5;
    const int lane = threadIdx.x & 31;
    const int tile = blockIdx.x * 8 + wid;
    const int mt   = tile & 7;          // 128/16 = 8 M-tiles (fastest)
    const int nt   = tile >> 3;
    const int m0   = mt * 16;
    const int f0   = nt * 16;
    const int row  = lane & 15;
    const int hi   = lane >> 4;         // 0: K pair {0,1}, 1: K pair {2,3}

    const float* wrow = w1 + (m0 + row) * D_CH;
    v8f acc = {0.f, 0.f, 0.f, 0.f, 0.f, 0.f, 0.f, 0.f};

#pragma unroll 8
    for (int k0 = 0; k0 < D_CH; k0 += 4) {
        const int ka = k0 + 2 * hi;
        v2f a, b;
        a.x = wrow[ka];
        a.y = wrow[ka + 1];
        b.x = fmaxf(cp[ka       * NFRM + f0 + row], 0.0f);
        b.y = fmaxf(cp[(ka + 1) * NFRM + f0 + row], 0.0f);
        acc = __builtin_amdgcn_wmma_f32_16x16x4_f32(false, a, false, b,
                                                    (short)0, acc, false, false);
    }

    const int mb = m0 + 8 * hi;
    const int n  = f0 + row;
#pragma unroll
    for (int r = 0; r < 8; ++r)
        h[(mb + r) * NFRM + n] = acc[r];
}

// ---------------------------------------------------------------------------
// 2a) Local IIR per (channel, chunk):  y[t] = dec*(h[t] + y[t-1]),  y(-1)=0.
//     Records chunk-end value E for the carry scan.
// ---------------------------------------------------------------------------
__global__ void k_scan(const float* __restrict__ dec,
                       const float* __restrict__ h,
                       float* __restrict__ y,
                       float* __restrict__ E) {
    const int tid = blockIdx.x * blockDim.x + threadIdx.x;   // D_CH*NCHUNK
    const int c   = tid >> 8;            // / NCHUNK
    const int k   = tid & (NCHUNK - 1);
    const float d = dec[c];
    const int base = c * NFRM + k * CHUNK;
    float yv = 0.0f;
    for (int t = 0; t < CHUNK; ++t) {
        yv = d * (h[base + t] + yv);
        y[base + t] = yv;
    }
    E[c * NCHUNK + k] = yv;
}

// ---------------------------------------------------------------------------
// 2b) Sequential carry scan over chunks (tiny: 128 ch x 256 chunks).
//     carry[k] = true y at end of chunk k-1;  carry' = E + carry*dec^CHUNK.
// ---------------------------------------------------------------------------
__global__ void k_carry(const float* __restrict__ dec,
                        const float* __restrict__ E,
                        float* __restrict__ carry) {
    const int c = threadIdx.x;
    const float d = dec[c];
    const float P = __powf(d, (float)CHUNK);
    float cy = 0.0f;
    for (int k = 0; k < NCHUNK; ++k) {
        carry[c * NCHUNK + k] = cy;
        cy = E[c * NCHUNK + k] + cy * P;
    }
}

// ---------------------------------------------------------------------------
// 2c) Fix-up:  y[c, k*CHUNK + t] += carry[c,k] * dec^(t+1).
//     y is laid out [c*NFRM + f] so the flat tid is the address.
// ---------------------------------------------------------------------------
__global__ void k_fix(const float* __restrict__ logdec,
                      const float* __restrict__ carry,
                      float* __restrict__ y) {
    const int tid = blockIdx.x * blockDim.x + threadIdx.x;   // D_CH*NFRM
    const int c = tid >> 16;             // / NFRM
    const int f = tid & (NFRM - 1);
    const int k = f >> 8;                // / CHUNK
    const int t = f & (CHUNK - 1);
    const float cr = carry[c * NCHUNK + k];
    if (cr != 0.0f)
        y[tid] += cr * __expf(logdec[c] * (float)(t + 1));
}

// ---------------------------------------------------------------------------
// 3) z = w2 @ y + h ;  cp_out = tanh(z * gain).  Accumulator seeded with h.
// ---------------------------------------------------------------------------
__global__ void k_gemm2(const float* __restrict__ y,
                        const float* __restrict__ w2,
                        const float* __restrict__ h,
                        const float* __restrict__ gain,
                        float* __restrict__ cpout) {
    const int wid  = threadIdx.x >> 5;
    const int lane = threadIdx.x & 31;
    const int tile = blockIdx.x * 8 + wid;
    const int mt   = tile & 7;
    const int nt   = tile >> 3;
    const int m0   = mt * 16;
    const int f0   = nt * 16;
    const int row  = lane & 15;
    const int hi   = lane >> 4;

    const int mb = m0 + 8 * hi;
    const int n  = f0 + row;

    v8f acc;
#pragma unroll
    for (int r = 0; r < 8; ++r)
        acc[r] = h[(mb + r) * NFRM + n];           // residual (orig)

    const float* wrow = w2 + (m0 + row) * D_CH;
#pragma unroll 8
    for (int k0 = 0; k0 < D_CH; k0 += 4) {
        const int ka = k0 + 2 * hi;
        v2f a, b;
        a.x = wrow[ka];
        a.y = wrow[ka + 1];
        b.x = y[ka       * NFRM + f0 + row];
        b.y = y[(ka + 1) * NFRM + f0 + row];
        acc = __builtin_amdgcn_wmma_f32_16x16x4_f32(false, a, false, b,
                                                    (short)0, acc, false, false);
    }

#pragma unroll
    for (int r = 0; r < 8; ++r) {
        const float g = gain[mb + r];
        cpout[(mb + r) * NFRM + n] = tanhf(acc[r] * g);
    }
}

// ---------------------------------------------------------------------------
// 4) audio_out[f, w] = sum_c audio[w, c] * cp_out[c, f].  M=512, K=128, N=F.
//    Output frame-major; each lane stores 8 consecutive floats (2x b128).
// ---------------------------------------------------------------------------
__global__ void k_gemm3(const float* __restrict__ cpout,
                        const float* __restrict__ audio,
                        float* __restrict__ aout) {
    const int wid  = threadIdx.x >> 5;
    const int lane = threadIdx.x & 31;
    const int tile = blockIdx.x * 8 + wid;
    const int mt   = tile & 31;          // 512/16 = 32 M-tiles (fastest)
    const int nt   = tile >> 5;
    const int m0   = mt * 16;
    const int f0   = nt * 16;
    const int row  = lane & 15;
    const int hi   = lane >> 4;

    const float* arow = audio + (m0 + row) * D_CH;
    v8f acc = {0.f, 0.f, 0.f, 0.f, 0.f, 0.f, 0.f, 0.f};

#pragma unroll 8
    for (int k0 = 0; k0 < D_CH; k0 += 4) {
        const int ka = k0 + 2 * hi;
        v2f a, b;
        a.x = arow[ka];
        a.y = arow[ka + 1];
        b.x = cpout[ka       * NFRM + f0 + row];
        b.y = cpout[(ka + 1) * NFRM + f0 + row];
        acc = __builtin_amdgcn_wmma_f32_16x16x4_f32(false, a, false, b,
                                                    (short)0, acc, false, false);
    }

    const int mb = m0 + 8 * hi;
    const int n  = f0 + row;
    float* o = aout + (size_t)n * NWIN + mb;
    v4f lo = {acc[0], acc[1], acc[2], acc[3]};
    v4f hv = {acc[4], acc[5], acc[6], acc[7]};
    *(v4f*)o       = lo;
    *(v4f*)(o + 4) = hv;
}

// ---------------------------------------------------------------------------
// Launch: all on `stream`, deterministic, scratch in d_ws only.
// ws float layout: dec[128] | logdec[128] | gain[128] | E[32768] |
//                  carry[32768] | h[128*65536] | y[128*65536]   (~67.4 MB)
// ---------------------------------------------------------------------------
extern "C" void kernel_launch(void* const* d_in, const int* in_sizes, int n_in,
                              void* d_out, int out_size, void* d_ws, size_t ws_size,
                              hipStream_t stream) {
    const float* cp     = (const float*)d_in[0];
    const float* w1     = (const float*)d_in[1];
    const float* w2     = (const float*)d_in[2];
    const float* audio  = (const float*)d_in[3];
    const float* decays = (const float*)d_in[4];
    const float* gains  = (const float*)d_in[5];

    float* out   = (float*)d_out;
    float* aout  = out;                                    // (F, 512) frame-major
    float* cpout = out + (size_t)NFRM * NWIN;              // (128, F)

    float* ws     = (float*)d_ws;
    float* dec    = ws;
    float* logdec = ws + D_CH;
    float* gain   = ws + 2 * D_CH;
    float* E      = ws + 3 * D_CH;
    float* carry  = E + D_CH * NCHUNK;
    float* h      = carry + D_CH * NCHUNK;
    float* y      = h + (size_t)D_CH * NFRM;

    k_prep <<<1, 128, 0, stream>>>(decays, gains, ws);
    k_gemm1<<<(8 * (NFRM / 16)) / 8, 256, 0, stream>>>(cp, w1, h);
    k_scan <<<(D_CH * NCHUNK) / 256, 256, 0, stream>>>(dec, h, y, E);
    k_carry<<<1, 128, 0, stream>>>(dec, E, carry);
    k_fix  <<<(D_CH * NFRM) / 256, 256, 0, stream>>>(logdec, carry, y);
    k_gemm2<<<(8 * (NFRM / 16)) / 8, 256, 0, stream>>>(y, w2, h, gain, cpout);
    k_gemm3<<<(32 * (NFRM / 16)) / 8, 256, 0, stream>>>(cpout, audio, aout);
}